// BackMapLayerWithSidechains_57019985821996
// MI455X (gfx1250) — compile-verified
//
#include <hip/hip_runtime.h>
#include <math.h>

#if defined(__AMDGCN__) && __has_builtin(__builtin_amdgcn_global_load_async_to_lds_b128) && __has_builtin(__builtin_amdgcn_s_wait_asynccnt)
#define USE_ASYNC_LDS 1
#else
#define USE_ASYNC_LDS 0
#endif

namespace {

constexpr int   kNres  = 50;
constexpr int   kV     = 4;
constexpr int   kNB    = 3 * kNres;          // 150 backbone atoms
constexpr int   kNS    = kNres * (kV + 1);   // 250 side-chain atoms
constexpr int   kAtoms = kNB + kNS;          // 400 atoms per batch element
constexpr float kChi0  = 2.0943951023931953f;
constexpr int   kTPB   = 256;

struct V3 { float x, y, z; };

__device__ __forceinline__ V3 v3sub(V3 a, V3 b) { return {a.x - b.x, a.y - b.y, a.z - b.z}; }

__device__ __forceinline__ V3 v3cross(V3 a, V3 b) {
    return { a.y * b.z - a.z * b.y,
             a.z * b.x - a.x * b.z,
             a.x * b.y - a.y * b.x };
}

__device__ __forceinline__ V3 v3unit(V3 a) {
    float inv = 1.0f / sqrtf(a.x * a.x + a.y * a.y + a.z * a.z);
    return { a.x * inv, a.y * inv, a.z * inv };
}

// NeRF placement with trig already evaluated (keeps the serial chain short).
__device__ __forceinline__ V3 nerf_place_tr(V3 A, V3 B, V3 C, float d,
                                            float st, float ct, float sp, float cp) {
    V3 bc = v3unit(v3sub(C, B));
    V3 n  = v3unit(v3cross(v3sub(B, A), bc));
    V3 m  = v3cross(n, bc);
    float wb = -d * ct;
    float wm =  d * st * cp;
    float wn =  d * st * sp;
    return { C.x + wb * bc.x + wm * m.x + wn * n.x,
             C.y + wb * bc.y + wm * m.y + wn * n.y,
             C.z + wb * bc.z + wm * m.z + wn * n.z };
}

__device__ __forceinline__ V3 nerf_place(V3 A, V3 B, V3 C, float d, float th, float ph) {
    float st, ct, sp, cp;
    sincosf(th, &st, &ct);
    sincosf(ph, &sp, &cp);
    return nerf_place_tr(A, B, C, d, st, ct, sp, cp);
}

__device__ __forceinline__ void store_atom(float* __restrict__ ob, int atom, V3 p) {
    ob[atom * 3 + 0] = p.x;
    ob[atom * 3 + 1] = p.y;
    ob[atom * 3 + 2] = p.z;
}

__device__ __forceinline__ V3 load_atom(const float* __restrict__ ob, int atom) {
    return { ob[atom * 3 + 0], ob[atom * 3 + 1], ob[atom * 3 + 2] };
}

#if USE_ASYNC_LDS
typedef int v4i_t __attribute__((ext_vector_type(4)));
typedef __attribute__((address_space(1))) v4i_t* gv4i_ptr;
typedef __attribute__((address_space(3))) v4i_t* lv4i_ptr;

// gfx1250 async bulk copy: 16 bytes global -> LDS, tracked by ASYNCcnt.
// Builtin signature (from compiler diagnostic): (v4i as1*, v4i as3*, imm, imm).
__device__ __forceinline__ void async_copy_b128(const float* gsrc, float* lds_dst) {
    __builtin_amdgcn_global_load_async_to_lds_b128(
        (gv4i_ptr)const_cast<float*>(gsrc),
        (lv4i_ptr)lds_dst,
        0, 0);
}
#endif

// ---------------------------------------------------------------------------
// Pass 0 (fully parallel, coalesced, bandwidth-bound): hoist all backbone trig
// off the serial recurrence. ws layout (floats):
//   [0,         batch*148)  sin(central_angles)
//   [batch*148, batch*296)  cos(central_angles)
//   [batch*296, batch*443)  sin(central_dihedrals)
//   [batch*443, batch*590)  cos(central_dihedrals)
// ---------------------------------------------------------------------------
__global__ void __launch_bounds__(kTPB)
trig_kernel(const float* __restrict__ ca, const float* __restrict__ ct,
            float* __restrict__ ws, int batch) {
    size_t nA = (size_t)batch * (kNB - 2);
    size_t nT = (size_t)batch * (kNB - 3);
    size_t idx = (size_t)blockIdx.x * blockDim.x + threadIdx.x;
    if (idx < nA) {
        float s, c;
        sincosf(ca[idx], &s, &c);
        ws[idx]      = s;
        ws[nA + idx] = c;
    } else if (idx < nA + nT) {
        size_t j = idx - nA;
        float s, c;
        sincosf(ct[j], &s, &c);
        ws[2 * nA + j]      = s;
        ws[2 * nA + nT + j] = c;
    }
}

// One lane per batch element: serial NeRF chain with precomputed trig.
__global__ void __launch_bounds__(kTPB)
backbone_pre_kernel(const float* __restrict__ cd, const float* __restrict__ ws,
                    float* __restrict__ out, int batch) {
    int b = blockIdx.x * blockDim.x + threadIdx.x;
    if (b >= batch) return;

    size_t nA = (size_t)batch * (kNB - 2);
    size_t nT = (size_t)batch * (kNB - 3);

    const float* __restrict__ cdb = cd + (size_t)b * (kNB - 1);
    const float* __restrict__ sA  = ws + (size_t)b * (kNB - 2);
    const float* __restrict__ cA  = ws + nA + (size_t)b * (kNB - 2);
    const float* __restrict__ sT  = ws + 2 * nA + (size_t)b * (kNB - 3);
    const float* __restrict__ cT  = ws + 2 * nA + nT + (size_t)b * (kNB - 3);
    float* __restrict__ ob = out + (size_t)b * kAtoms * 3;

    __builtin_prefetch(cdb, 0, 1);
    __builtin_prefetch(sA, 0, 1);
    __builtin_prefetch(cA, 0, 1);
    __builtin_prefetch(sT, 0, 1);
    __builtin_prefetch(cT, 0, 1);

    V3 A = {0.0f, 0.0f, 0.0f};
    float d0 = cdb[0];
    V3 B = {d0, 0.0f, 0.0f};
    float d1 = cdb[1];
    V3 C = {d0 - d1 * cA[0], d1 * sA[0], 0.0f};

    store_atom(ob, 0, A);
    store_atom(ob, 1, B);
    store_atom(ob, 2, C);

    #pragma unroll 4
    for (int i = 0; i < kNB - 3; ++i) {
        if ((i & 31) == 0) {
            __builtin_prefetch(cdb + i + 34, 0, 1);
            __builtin_prefetch(sA + i + 33, 0, 1);
            __builtin_prefetch(cA + i + 33, 0, 1);
            __builtin_prefetch(sT + i + 32, 0, 1);
            __builtin_prefetch(cT + i + 32, 0, 1);
        }
        V3 D = nerf_place_tr(A, B, C, cdb[2 + i], sA[1 + i], cA[1 + i], sT[i], cT[i]);
        store_atom(ob, 3 + i, D);
        A = B; B = C; C = D;
    }
}

// Fallback (no workspace): fused sincos in the chain.
__global__ void __launch_bounds__(kTPB)
backbone_kernel(const float* __restrict__ cd, const float* __restrict__ ca,
                const float* __restrict__ ct, float* __restrict__ out, int batch) {
    int b = blockIdx.x * blockDim.x + threadIdx.x;
    if (b >= batch) return;

    const float* __restrict__ cdb = cd + (size_t)b * (kNB - 1);
    const float* __restrict__ cab = ca + (size_t)b * (kNB - 2);
    const float* __restrict__ ctb = ct + (size_t)b * (kNB - 3);
    float* __restrict__ ob = out + (size_t)b * kAtoms * 3;

    V3 A = {0.0f, 0.0f, 0.0f};
    float d0 = cdb[0];
    V3 B = {d0, 0.0f, 0.0f};
    float s0, c0;
    sincosf(cab[0], &s0, &c0);
    float d1 = cdb[1];
    V3 C = {d0 - d1 * c0, d1 * s0, 0.0f};

    store_atom(ob, 0, A);
    store_atom(ob, 1, B);
    store_atom(ob, 2, C);

    #pragma unroll 4
    for (int i = 0; i < kNB - 3; ++i) {
        V3 D = nerf_place(A, B, C, cdb[2 + i], cab[1 + i], ctb[i]);
        store_atom(ob, 3 + i, D);
        A = B; B = C; C = D;
    }
}

// One lane per (batch, residue): 5 chained NeRF steps off that residue's frame.
// A block's 256 chains read three CONTIGUOUS global regions -> stage them into
// LDS with gfx1250 async b128 copies (ASYNCcnt), then read per-lane from LDS.
__global__ void __launch_bounds__(kTPB)
sidechain_kernel(const float* __restrict__ sd, const float* __restrict__ sa,
                 const float* __restrict__ sdh, float* __restrict__ out, int batch) {
    const int total      = batch * kNres;
    const int block_base = blockIdx.x * kTPB;
    const int t          = threadIdx.x;
    const int idx        = block_base + t;

#if USE_ASYNC_LDS
    __shared__ alignas(16) float s_sd[kTPB * (kV + 1)];   // 1280 floats
    __shared__ alignas(16) float s_sa[kTPB * (kV + 1)];   // 1280 floats
    __shared__ alignas(16) float s_st[kTPB * kV];         // 1024 floats
    const bool full = (block_base + kTPB) <= total;
    if (full) {
        const float* g_sd = sd  + (size_t)block_base * (kV + 1);
        const float* g_sa = sa  + (size_t)block_base * (kV + 1);
        const float* g_st = sdh + (size_t)block_base * kV;
        // sd/sa: 320 float4 each; st: 256 float4. 16B-aligned block bases.
        async_copy_b128(g_sd + 4 * t, s_sd + 4 * t);
        async_copy_b128(g_sa + 4 * t, s_sa + 4 * t);
        async_copy_b128(g_st + 4 * t, s_st + 4 * t);
        if (t < (kTPB * (kV + 1) / 4 - kTPB)) {   // remaining 64 float4 of sd/sa
            async_copy_b128(g_sd + 4 * (kTPB + t), s_sd + 4 * (kTPB + t));
            async_copy_b128(g_sa + 4 * (kTPB + t), s_sa + 4 * (kTPB + t));
        }
        __builtin_amdgcn_s_wait_asynccnt(0);
        __syncthreads();
    }
#endif

    if (idx >= total) return;
    const int r = idx % kNres;
    const int b = idx / kNres;
    float* __restrict__ ob = out + (size_t)b * kAtoms * 3;

    // Carry starts at (C, N, CA) of this residue (written by the backbone pass).
    V3 A = load_atom(ob, 3 * r + 2);
    V3 B = load_atom(ob, 3 * r + 0);
    V3 C = load_atom(ob, 3 * r + 1);

#if USE_ASYNC_LDS
    if (full) {
        const float* __restrict__ lsd = s_sd + t * (kV + 1);
        const float* __restrict__ lsa = s_sa + t * (kV + 1);
        const float* __restrict__ lst = s_st + t * kV;
        #pragma unroll
        for (int k = 0; k <= kV; ++k) {
            float ph = (k == 0) ? kChi0 : lst[k - 1];
            V3 D = nerf_place(A, B, C, lsd[k], lsa[k], ph);
            store_atom(ob, kNB + r * (kV + 1) + k, D);
            A = B; B = C; C = D;
        }
        return;
    }
#endif

    const float* __restrict__ sdb = sd  + (size_t)idx * (kV + 1);
    const float* __restrict__ sab = sa  + (size_t)idx * (kV + 1);
    const float* __restrict__ stb = sdh + (size_t)idx * kV;
    #pragma unroll
    for (int k = 0; k <= kV; ++k) {
        float ph = (k == 0) ? kChi0 : stb[k - 1];
        V3 D = nerf_place(A, B, C, sdb[k], sab[k], ph);
        store_atom(ob, kNB + r * (kV + 1) + k, D);
        A = B; B = C; C = D;
    }
}

} // anonymous namespace

extern "C" void kernel_launch(void* const* d_in, const int* in_sizes, int n_in,
                              void* d_out, int out_size, void* d_ws, size_t ws_size,
                              hipStream_t stream) {
    const float* cd  = (const float*)d_in[0];   // central_distances  (B, NB-1)
    const float* ca  = (const float*)d_in[1];   // central_angles     (B, NB-2)
    const float* ct  = (const float*)d_in[2];   // central_dihedrals  (B, NB-3)
    const float* sd  = (const float*)d_in[3];   // side_distances     (B, NS)
    const float* sa  = (const float*)d_in[4];   // side_angles        (B, NS)
    const float* sdh = (const float*)d_in[5];   // side_dihedrals     (B, N_RES*V)
    float* out = (float*)d_out;
    float* ws  = (float*)d_ws;

    const int batch = in_sizes[0] / (kNB - 1);

    const size_t trig_bytes =
        (size_t)batch * (2 * (kNB - 2) + 2 * (kNB - 3)) * sizeof(float);

    if (ws_size >= trig_bytes) {
        const size_t total = (size_t)batch * ((kNB - 2) + (kNB - 3));
        const int blocks = (int)((total + kTPB - 1) / kTPB);
        trig_kernel<<<blocks, kTPB, 0, stream>>>(ca, ct, ws, batch);

        const int bblocks = (batch + kTPB - 1) / kTPB;
        backbone_pre_kernel<<<bblocks, kTPB, 0, stream>>>(cd, ws, out, batch);
    } else {
        const int bblocks = (batch + kTPB - 1) / kTPB;
        backbone_kernel<<<bblocks, kTPB, 0, stream>>>(cd, ca, ct, out, batch);
    }

    {
        const int total  = batch * kNres;
        const int blocks = (total + kTPB - 1) / kTPB;
        sidechain_kernel<<<blocks, kTPB, 0, stream>>>(sd, sa, sdh, out, batch);
    }
}